// PathDevelopmentNetwork_10110353015092
// MI455X (gfx1250) — compile-verified
//
#include <hip/hip_runtime.h>

typedef float v2f __attribute__((ext_vector_type(2)));
typedef float v8f __attribute__((ext_vector_type(8)));

constexpr int kB  = 8;
constexpr int kT  = 2048;
constexpr int kD  = 64;
constexpr int kL  = 64;            // time steps per chunk
constexpr int kNC = kT / kL;       // 32 chunks -> scan depth 64 + 31
constexpr int kXP = 76;            // LDS pitch for X/P (conflict-free A-frag b64 loads)
constexpr int kPP = 160;           // row-pair pitch for M (%64==32 -> conflict-free b64 B loads)
constexpr int kTS = 8;             // time steps per block in pass 3

__device__ __forceinline__ v8f wmma4(const v2f a, const v2f b, const v8f c) {
  return __builtin_amdgcn_wmma_f32_16x16x4_f32(false, a, false, b, (short)0, c,
                                               false, false);
}

// Load the 16 A-fragments (16x4 slabs) of one 16-row band from a row-major LDS matrix.
// 32-bit A 16x4 layout: lanes 0-15 hold M=0..15; VGPR v, half h -> K = 2*h + v.
__device__ __forceinline__ void load_a_frags(v2f a[16], const float* Xm, int pitch,
                                             int ti, int half, int lrow) {
  const int base = (16 * ti + lrow) * pitch + 2 * half;
#pragma unroll
  for (int k = 0; k < 16; ++k) a[k] = *(const v2f*)(Xm + base + 4 * k);
}

// 16x16 output tile, B in row-pair-interleaved LDS layout: (r,c) at (r>>1)*kPP + 2c + (r&1).
// B frag rows 4k+2h, 4k+2h+1 -> single b64 at pair-row (2k+h). Dual acc chains.
__device__ __forceinline__ v8f mul_tile_paired(const v2f a[16], const float* Bm,
                                               int tj, int half, int lrow) {
  v8f acc0 = {}, acc1 = {};
  const int col2 = 2 * (16 * tj + lrow);
#pragma unroll
  for (int k = 0; k < 16; k += 2) {
    const v2f b0 = *(const v2f*)(Bm + (2 * k + half) * kPP + col2);
    const v2f b1 = *(const v2f*)(Bm + (2 * (k + 1) + half) * kPP + col2);
    acc0 = wmma4(a[k], b0, acc0);
    acc1 = wmma4(a[k + 1], b1, acc1);
  }
  return acc0 + acc1;
}

// 16x16 output tile, B row-major (pitch kD) straight from global. Dual acc chains.
__device__ __forceinline__ v8f mul_tile_g(const v2f a[16], const float* Bm,
                                          int tj, int half, int lrow) {
  v8f acc0 = {}, acc1 = {};
  const int col = 16 * tj + lrow;
#pragma unroll
  for (int k = 0; k < 16; k += 2) {
    const int r0 = 4 * k + 2 * half;
    const int r1 = 4 * (k + 1) + 2 * half;
    v2f b0, b1;
    b0.x = Bm[r0 * kD + col];
    b0.y = Bm[(r0 + 1) * kD + col];
    b1.x = Bm[r1 * kD + col];
    b1.y = Bm[(r1 + 1) * kD + col];
    acc0 = wmma4(a[k], b0, acc0);
    acc1 = wmma4(a[k + 1], b1, acc1);
  }
  return acc0 + acc1;
}

// Pass 1: per (batch, chunk) local cumulative products + chunk total.
__global__ __launch_bounds__(512) void pdev_pass1(
    const float* __restrict__ Z, const float* __restrict__ A1,
    const float* __restrict__ A2, float* __restrict__ out,
    float* __restrict__ G) {
  extern __shared__ float lds[];
  float* sM = lds;                   // 32 * kPP  (row-pair interleaved M)
  float* sX = sM + 32 * kPP;         // 64 * kXP  (row-major running product)

  const int b = blockIdx.x / kNC;
  const int c = blockIdx.x % kNC;
  const int tid  = threadIdx.x;
  const int lane = tid & 31, wave = tid >> 5;
  const int half = lane >> 4, lrow = lane & 15;
  const int ti = wave >> 2, tj = wave & 3;   // 16 waves -> 4x4 tile grid

  // This thread owns rows {2pr, 2pr+1} x cols {4cg..4cg+3} of M (pairs for b64 stores).
  const int pr = tid >> 4;           // 0..31 pair-row
  const int cg = tid & 15;           // col group
  const int r0 = 2 * pr, r1 = 2 * pr + 1;
  float rS1e[4], rS1o[4], rS2e[4], rS2o[4];  // skew-symmetrized params in registers
#pragma unroll
  for (int j = 0; j < 4; ++j) {
    const int cc = 4 * cg + j;
    rS1e[j] = A1[r0 * kD + cc] - A1[cc * kD + r0];
    rS1o[j] = A1[r1 * kD + cc] - A1[cc * kD + r1];
    rS2e[j] = A2[r0 * kD + cc] - A2[cc * kD + r0];
    rS2o[j] = A2[r1 * kD + cc] - A2[cc * kD + r1];
  }

  // X = I in LDS; emit local prefix at chunk start (= I).
  for (int e = tid; e < kD * kD; e += 512) {
    const int r = e >> 6, col = e & 63;
    const float idv = (r == col) ? 1.0f : 0.0f;
    sX[r * kXP + col] = idv;
    out[((size_t)(b * kT + c * kL)) * (kD * kD) + e] = idv;
  }
  __syncthreads();

  const int base    = c * kL;
  const int sBridge = base + kL;                              // bridging matrix
  const int sStop   = (c == kNC - 1) ? (sBridge - 1) : sBridge;
  for (int s = base + 1; s <= sStop; ++s) {
    const float z0 = Z[((size_t)(b * kT + s)) * 2 + 0];
    const float z1 = Z[((size_t)(b * kT + s)) * 2 + 1];
    // Build M_s in row-pair layout: 8 FMAs + 4 ds_store_b64 per thread.
#pragma unroll
    for (int j = 0; j < 4; ++j) {
      v2f m;
      m.x = z0 * rS1e[j] + z1 * rS2e[j];
      m.y = z0 * rS1o[j] + z1 * rS2o[j];
      *(v2f*)(sM + pr * kPP + 2 * (4 * cg + j)) = m;
    }
    v2f a[16];
    load_a_frags(a, sX, kXP, ti, half, lrow);  // read old X before it is overwritten
    __syncthreads();                            // M visible; X reads complete
    const v8f acc = mul_tile_paired(a, sM, tj, half, lrow);
    if (s == sBridge) {                         // chunk total -> workspace
      float* g = G + ((size_t)(b * (kNC - 1) + c)) * (kD * kD);
#pragma unroll
      for (int v = 0; v < 8; ++v) {
        const int r = 16 * ti + v + 8 * half, col = 16 * tj + lrow;
        g[r * kD + col] = acc[v];
      }
    } else {                                    // local prefix -> out + LDS
      float* o = out + ((size_t)(b * kT + s)) * (kD * kD);
#pragma unroll
      for (int v = 0; v < 8; ++v) {
        const int r = 16 * ti + v + 8 * half, col = 16 * tj + lrow;
        sX[r * kXP + col] = acc[v];
        o[r * kD + col] = acc[v];
      }
    }
    __syncthreads();                            // X/M stable for next step
  }
}

// Pass 2: exclusive scan of chunk totals per batch (chain of 31 matmuls).
__global__ __launch_bounds__(512) void pdev_pass2(
    const float* __restrict__ G, float* __restrict__ P) {
  extern __shared__ float lds[];
  float* sM = lds;                   // 32 * kPP (row-pair interleaved)
  float* sX = sM + 32 * kPP;         // 64 * kXP
  const int b = blockIdx.x;
  const int tid  = threadIdx.x;
  const int lane = tid & 31, wave = tid >> 5;
  const int half = lane >> 4, lrow = lane & 15;
  const int ti = wave >> 2, tj = wave & 3;

  for (int e = tid; e < kD * kD; e += 512) {
    const int r = e >> 6, col = e & 63;
    sX[r * kXP + col] = (r == col) ? 1.0f : 0.0f;
  }
  __syncthreads();
  for (int c = 1; c < kNC; ++c) {
    const float* g = G + ((size_t)(b * (kNC - 1) + (c - 1))) * (kD * kD);
    for (int e = tid; e < kD * kD; e += 512) {
      const int r = e >> 6, col = e & 63;   // transpose into row-pair layout
      sM[(r >> 1) * kPP + 2 * col + (r & 1)] = g[e];
    }
    v2f a[16];
    load_a_frags(a, sX, kXP, ti, half, lrow);
    __syncthreads();
    const v8f acc = mul_tile_paired(a, sM, tj, half, lrow);
    float* p = P + ((size_t)(b * kNC + c)) * (kD * kD);
#pragma unroll
    for (int v = 0; v < 8; ++v) {
      const int r = 16 * ti + v + 8 * half, col = 16 * tj + lrow;
      sX[r * kXP + col] = acc[v];
      p[r * kD + col] = acc[v];
    }
    __syncthreads();
  }
}

// Pass 3: out[b,t] = P[b,c] @ out[b,t] in place (chunks c >= 1), fully parallel.
__global__ __launch_bounds__(512) void pdev_pass3(
    const float* __restrict__ P, float* __restrict__ out) {
  extern __shared__ float lds[];
  float* sP = lds;                   // 64 * kXP
  constexpr int kPer = kL / kTS;     // 8 blocks per chunk
  const int gid = blockIdx.x;
  const int b   = gid / ((kNC - 1) * kPer);
  const int rem = gid % ((kNC - 1) * kPer);
  const int c   = 1 + rem / kPer;
  const int seg = rem % kPer;
  const int t0  = c * kL + seg * kTS;

  const int tid  = threadIdx.x;
  const int lane = tid & 31, wave = tid >> 5;
  const int half = lane >> 4, lrow = lane & 15;
  const int ti = wave >> 2, tj = wave & 3;

  const float* p = P + ((size_t)(b * kNC + c)) * (kD * kD);
  for (int e = tid; e < kD * kD; e += 512)
    sP[(e >> 6) * kXP + (e & 63)] = p[e];
  __syncthreads();
  v2f a[16];
  load_a_frags(a, sP, kXP, ti, half, lrow);    // A = prefix, constant across t

  for (int t = t0; t < t0 + kTS; ++t) {
    float* x = out + ((size_t)(b * kT + t)) * (kD * kD);
    const v8f acc = mul_tile_g(a, x, tj, half, lrow);  // B read from global
    __syncthreads();                // all reads of X_t done before any writes
#pragma unroll
    for (int v = 0; v < 8; ++v) {
      const int r = 16 * ti + v + 8 * half, col = 16 * tj + lrow;
      x[r * kD + col] = acc[v];
    }
  }
}

extern "C" void kernel_launch(void* const* d_in, const int* in_sizes, int n_in,
                              void* d_out, int out_size, void* d_ws, size_t ws_size,
                              hipStream_t stream) {
  (void)in_sizes; (void)n_in; (void)out_size; (void)ws_size;
  const float* Z  = (const float*)d_in[0];
  const float* A1 = (const float*)d_in[1];
  const float* A2 = (const float*)d_in[2];
  float* out = (float*)d_out;
  float* G = (float*)d_ws;                              // [B][NC-1][64*64] ~4 MB
  float* P = G + (size_t)kB * (kNC - 1) * kD * kD;      // [B][NC][64*64]  ~4 MB

  constexpr size_t lds12 = (size_t)(32 * kPP + kD * kXP) * sizeof(float); // 39,936 B
  constexpr size_t lds3  = (size_t)(kD * kXP) * sizeof(float);            // 19,456 B

  pdev_pass1<<<kB * kNC, 512, lds12, stream>>>(Z, A1, A2, out, G);
  pdev_pass2<<<kB, 512, lds12, stream>>>(G, P);
  pdev_pass3<<<kB * (kNC - 1) * (kL / kTS), 512, lds3, stream>>>(P, out);
}